// HistogramQuantizer_1614907703439
// MI455X (gfx1250) — compile-verified
//
#include <hip/hip_runtime.h>
#include <cmath>
#include <stdint.h>

// ---------------------------------------------------------------------------
// HistogramQuantizer for MI455X (gfx1250, wave32).
// Percentile via radix-select on abs-f32 bit patterns instead of a full sort.
// Streaming passes use TDM tensor_load_to_lds (TENSORcnt) double buffering.
// ---------------------------------------------------------------------------

#define BLOCK       256
#define TILE_ELEMS  4096          // 16 KB per LDS tile, x2 buffers + 16KB hist = 48KB LDS
#define HIST_BINS   4096          // top 12 bits of abs(f32) pattern
#define QMAX        127.0f
#define COLLECT_MAX (1u << 20)

#if defined(__has_builtin)
#  if __has_builtin(__builtin_amdgcn_tensor_load_to_lds)
#    define HAVE_TDM 1
#  endif
#  if __has_builtin(__builtin_amdgcn_s_wait_tensorcnt)
#    define HAVE_TCNT 1
#  endif
#endif

typedef __attribute__((ext_vector_type(4))) unsigned int v4u;
typedef __attribute__((ext_vector_type(8))) int          v8i;
typedef __attribute__((ext_vector_type(4))) int          v4i;

#if defined(HAVE_TDM)
// Issue a 1D TDM copy of `elems` f32 from global `gaddr` to LDS offset `lds_off`.
// D# layout per CDNA5 ISA ch.8: group0 = {flags, lds_addr, global_addr, type=2},
// group1 = {data_size=4B, tensor_dim0, tile_dim0, stride0}; groups 2/3 zero (<=2D).
// clang-23 toolchain: 6-arg builtin (v4u, v8i, v4i, v4i, v8i, i32 cpol).
__device__ __forceinline__ void tdm_load_1d(uint32_t lds_off, uint64_t gaddr, uint32_t elems) {
  v4u g0;
  g0.x = 1u;                                           // count=1 (valid), user mode
  g0.y = lds_off;                                      // LDS byte address
  g0.z = (uint32_t)(gaddr & 0xFFFFFFFFull);            // global_addr[31:0]
  g0.w = ((uint32_t)(gaddr >> 32) & 0x01FFFFFFu)       // global_addr[56:32]
         | 0x80000000u;                                // type=2 ("image")
  v8i g1;
  g1[0] = (int)(2u << 16);                             // data_size=4B, workgroup_mask=0
  g1[1] = (int)((elems & 0xFFFFu) << 16);              // tensor_dim0[15:0]
  g1[2] = (int)((elems >> 16) & 0xFFFFu);              // tensor_dim0[31:16], tensor_dim1=0
  g1[3] = (int)(elems << 16);                          // tile_dim0 = elems
  g1[4] = 0;                                           // tile_dim1 = tile_dim2 = 0
  g1[5] = (int)elems;                                  // tensor_dim0_stride (don't care, 1 row)
  g1[6] = 0;
  g1[7] = 0;
  v4i gz4 = {0, 0, 0, 0};
  v8i gz8 = {0, 0, 0, 0, 0, 0, 0, 0};
  __builtin_amdgcn_tensor_load_to_lds(g0, g1, gz4, gz4, gz8, 0);
}
__device__ __forceinline__ void tdm_wait_le1() {
#if defined(HAVE_TCNT)
  __builtin_amdgcn_s_wait_tensorcnt(1);
#else
  asm volatile("s_wait_tensorcnt 0x1" ::: "memory");
#endif
}
__device__ __forceinline__ void tdm_wait_0() {
#if defined(HAVE_TCNT)
  __builtin_amdgcn_s_wait_tensorcnt(0);
#else
  asm volatile("s_wait_tensorcnt 0x0" ::: "memory");
#endif
}
#endif // HAVE_TDM

// ---------------------------------------------------------------------------
__global__ void __launch_bounds__(BLOCK) init_kernel(uint32_t* maxbits, uint32_t* ccount,
                                                     uint32_t* selB, uint32_t* selR,
                                                     uint32_t* hist) {
  int i = blockIdx.x * blockDim.x + threadIdx.x;
  if (i == 0) { *maxbits = 0u; *ccount = 0u; *selB = 0u; *selR = 0u; }
  for (int j = i; j < HIST_BINS; j += gridDim.x * blockDim.x) hist[j] = 0u;
}

// ---------------------------------------------------------------------------
// Pass A: histogram of top-12 abs bits + global abs-max.
// TDM stages 16KB tiles into LDS (double buffered); wave0 drives the DMA.
__global__ void __launch_bounds__(BLOCK) histmax_kernel(const float* __restrict__ x,
                                                        long long n,
                                                        uint32_t* __restrict__ hist,
                                                        uint32_t* __restrict__ maxbits) {
  __shared__ __align__(16) float tile[2][TILE_ELEMS];
  __shared__ uint32_t lhist[HIST_BINS];
  __shared__ uint32_t smax;

  const int tid = threadIdx.x;
  for (int i = tid; i < HIST_BINS; i += BLOCK) lhist[i] = 0u;
  if (tid == 0) smax = 0u;
  __syncthreads();

  uint32_t lmax = 0u;
  const long long tilesTotal = n / TILE_ELEMS;
  const long long stride = gridDim.x;
  const bool wave0 = (tid < 32);
  int buf = 0;

#if defined(HAVE_TDM)
  if ((long long)blockIdx.x < tilesTotal && wave0) {
    tdm_load_1d((uint32_t)(uintptr_t)&tile[0][0],
                (uint64_t)(uintptr_t)(x + (long long)blockIdx.x * TILE_ELEMS), TILE_ELEMS);
  }
#endif

  for (long long t = blockIdx.x; t < tilesTotal; t += stride) {
    const long long tn = t + stride;
#if defined(HAVE_TDM)
    if (wave0) {
      if (tn < tilesTotal) {
        tdm_load_1d((uint32_t)(uintptr_t)&tile[buf ^ 1][0],
                    (uint64_t)(uintptr_t)(x + tn * TILE_ELEMS), TILE_ELEMS);
        tdm_wait_le1();           // oldest (tile t) done; tile tn still in flight
      } else {
        tdm_wait_0();
      }
    }
#else
    // Fallback: cooperative b128 copy into LDS.
    {
      const uint4* src = (const uint4*)(x + t * TILE_ELEMS);
      uint4* dst = (uint4*)&tile[buf][0];
      for (int j = tid; j < TILE_ELEMS / 4; j += BLOCK) dst[j] = src[j];
    }
#endif
    __syncthreads();              // tile[buf] visible to all waves
    const uint4* tp = (const uint4*)&tile[buf][0];
#pragma unroll
    for (int j = 0; j < (TILE_ELEMS / 4) / BLOCK; ++j) {
      uint4 v = tp[j * BLOCK + tid];
      uint32_t b0 = v.x & 0x7FFFFFFFu, b1 = v.y & 0x7FFFFFFFu;
      uint32_t b2 = v.z & 0x7FFFFFFFu, b3 = v.w & 0x7FFFFFFFu;
      lmax = max(max(lmax, b0), max(b1, max(b2, b3)));
      atomicAdd(&lhist[b0 >> 19], 1u);
      atomicAdd(&lhist[b1 >> 19], 1u);
      atomicAdd(&lhist[b2 >> 19], 1u);
      atomicAdd(&lhist[b3 >> 19], 1u);
    }
    __syncthreads();              // everyone done reading tile[buf] before reuse
    buf ^= 1;
  }

  // Tail (n % TILE_ELEMS) — plain loads, grid-strided.
  const long long tail0 = tilesTotal * TILE_ELEMS;
  for (long long i = tail0 + (long long)blockIdx.x * BLOCK + tid; i < n;
       i += (long long)gridDim.x * BLOCK) {
    uint32_t b = __float_as_uint(x[i]) & 0x7FFFFFFFu;
    lmax = max(lmax, b);
    atomicAdd(&lhist[b >> 19], 1u);
  }

  atomicMax(&smax, lmax);
  __syncthreads();
  for (int i = tid; i < HIST_BINS; i += BLOCK) {
    uint32_t c = lhist[i];
    if (c) atomicAdd(&hist[i], c);
  }
  if (tid == 0) atomicMax(maxbits, smax);
}

// ---------------------------------------------------------------------------
// Pass B: find bucket containing rank k and the residual rank inside it.
__global__ void scan_kernel(const uint32_t* __restrict__ hist, long long k,
                            uint32_t* selB, uint32_t* selR) {
  if (threadIdx.x == 0 && blockIdx.x == 0) {
    long long cum = 0;
    uint32_t b = HIST_BINS - 1, r = 0;
    for (int i = 0; i < HIST_BINS; ++i) {
      long long c = (long long)hist[i];
      if (cum + c > k) { b = (uint32_t)i; r = (uint32_t)(k - cum); break; }
      cum += c;
    }
    *selB = b; *selR = r;
  }
}

// ---------------------------------------------------------------------------
// Pass C (fused): streaming fake-quantize + collect abs-bit patterns belonging
// to the selected bucket. out = clip(rint(x*scale),±127)/scale, scale=127/max.
__device__ __forceinline__ float quant1(float v, float sc) {
  float q = rintf(v * sc);
  q = fminf(fmaxf(q, -QMAX), QMAX);
  return q / sc;
}

__global__ void __launch_bounds__(BLOCK) quant_collect_kernel(
    const float* __restrict__ x, float* __restrict__ out, long long n,
    const uint32_t* __restrict__ maxbits, const uint32_t* __restrict__ selB,
    uint32_t* __restrict__ ccount, uint32_t* __restrict__ collect, uint32_t cap) {
  __shared__ float s_scale;
  __shared__ uint32_t s_bucket;
  if (threadIdx.x == 0) {
    s_scale = QMAX / __uint_as_float(*maxbits);
    s_bucket = *selB;
  }
  __syncthreads();
  const float sc = s_scale;
  const uint32_t sb = s_bucket;

  const long long nv = n >> 2;
  const float4* xv = (const float4*)x;
  float4* ov = (float4*)out;
  const long long gstride = (long long)gridDim.x * BLOCK;

  for (long long i = (long long)blockIdx.x * BLOCK + threadIdx.x; i < nv; i += gstride) {
    float4 v = xv[i];
    float4 o;
    o.x = quant1(v.x, sc); o.y = quant1(v.y, sc);
    o.z = quant1(v.z, sc); o.w = quant1(v.w, sc);
    ov[i] = o;
    uint32_t b0 = __float_as_uint(v.x) & 0x7FFFFFFFu;
    uint32_t b1 = __float_as_uint(v.y) & 0x7FFFFFFFu;
    uint32_t b2 = __float_as_uint(v.z) & 0x7FFFFFFFu;
    uint32_t b3 = __float_as_uint(v.w) & 0x7FFFFFFFu;
    if ((b0 >> 19) == sb) { uint32_t j = atomicAdd(ccount, 1u); if (j < cap) collect[j] = b0; }
    if ((b1 >> 19) == sb) { uint32_t j = atomicAdd(ccount, 1u); if (j < cap) collect[j] = b1; }
    if ((b2 >> 19) == sb) { uint32_t j = atomicAdd(ccount, 1u); if (j < cap) collect[j] = b2; }
    if ((b3 >> 19) == sb) { uint32_t j = atomicAdd(ccount, 1u); if (j < cap) collect[j] = b3; }
  }
  for (long long i = (nv << 2) + (long long)blockIdx.x * BLOCK + threadIdx.x; i < n; i += gstride) {
    float v = x[i];
    out[i] = quant1(v, sc);
    uint32_t b = __float_as_uint(v) & 0x7FFFFFFFu;
    if ((b >> 19) == sb) { uint32_t j = atomicAdd(ccount, 1u); if (j < cap) collect[j] = b; }
  }
}

// ---------------------------------------------------------------------------
// Pass D: exact order statistic among collected candidates via counting select
// on the remaining 19 bits (8 + 8 + 3), then EMA buffer output.
__global__ void __launch_bounds__(BLOCK) select_kernel(
    const uint32_t* __restrict__ collect, const uint32_t* __restrict__ ccount,
    uint32_t cap, const uint32_t* __restrict__ selB, const uint32_t* __restrict__ selR,
    float* __restrict__ out_buffer) {
  __shared__ uint32_t cnt[256];
  __shared__ uint32_t s_known, s_rank;
  uint32_t m = *ccount; if (m > cap) m = cap;
  if (threadIdx.x == 0) { s_known = (*selB) << 19; s_rank = *selR; }
  __syncthreads();

  const int      shifts[3] = {11, 3, 0};
  const uint32_t widths[3] = {256u, 256u, 8u};
  const uint32_t pmask[3]  = {0x7FF80000u, 0x7FFFF800u, 0x7FFFFFF8u};

#pragma unroll
  for (int r = 0; r < 3; ++r) {
    for (int i = threadIdx.x; i < 256; i += BLOCK) cnt[i] = 0u;
    __syncthreads();
    const uint32_t kn = s_known;
    for (uint32_t i = threadIdx.x; i < m; i += BLOCK) {
      uint32_t b = collect[i];
      if ((b & pmask[r]) == kn) atomicAdd(&cnt[(b >> shifts[r]) & (widths[r] - 1u)], 1u);
    }
    __syncthreads();
    if (threadIdx.x == 0) {
      uint32_t rk = s_rank, cum = 0, dsel = widths[r] - 1u;
      for (uint32_t d = 0; d < widths[r]; ++d) {
        uint32_t c = cnt[d];
        if (cum + c > rk) { dsel = d; rk -= cum; break; }
        cum += c;
      }
      s_known = kn | (dsel << shifts[r]);
      s_rank = rk;
    }
    __syncthreads();
  }
  if (threadIdx.x == 0) {
    float clampv = (m == 0u) ? __uint_as_float((*selB) << 19) : __uint_as_float(s_known);
    *out_buffer = 100.0f * 0.99f + clampv * (1.0f - 0.99f);
  }
}

// ---------------------------------------------------------------------------
extern "C" void kernel_launch(void* const* d_in, const int* in_sizes, int n_in,
                              void* d_out, int out_size, void* d_ws, size_t ws_size,
                              hipStream_t stream) {
  const long long n = (long long)in_sizes[0];
  const float* x = (const float*)d_in[0];
  float* out = (float*)d_out;
  float* newbuf = out + n;                      // tuple output: (out[n], new_buffer)

  uint8_t* ws = (uint8_t*)d_ws;
  uint32_t* maxbits = (uint32_t*)(ws + 0);
  uint32_t* ccount  = (uint32_t*)(ws + 4);
  uint32_t* selB    = (uint32_t*)(ws + 8);
  uint32_t* selR    = (uint32_t*)(ws + 12);
  uint32_t* hist    = (uint32_t*)(ws + 16);
  uint32_t* collect = (uint32_t*)(ws + 16 + HIST_BINS * 4);
  const long long fixedBytes = 16 + HIST_BINS * 4;
  long long capll = ((long long)ws_size - fixedBytes) / 4;
  if (capll < 0) capll = 0;
  uint32_t cap = (capll > (long long)COLLECT_MAX) ? COLLECT_MAX : (uint32_t)capll;

  long long k = llround((99.99 / 100.0) * (double)n) - 1;   // rank of the percentile
  if (k < 0) k = 0;
  if (k >= n) k = n - 1;

  init_kernel<<<dim3(16), dim3(BLOCK), 0, stream>>>(maxbits, ccount, selB, selR, hist);

  long long tiles = n / TILE_ELEMS;
  int gB = 2048;
  if (tiles < gB) gB = (tiles > 0) ? (int)tiles : 1;
  histmax_kernel<<<dim3(gB), dim3(BLOCK), 0, stream>>>(x, n, hist, maxbits);

  scan_kernel<<<dim3(1), dim3(32), 0, stream>>>(hist, k, selB, selR);

  quant_collect_kernel<<<dim3(4096), dim3(BLOCK), 0, stream>>>(x, out, n, maxbits, selB,
                                                               ccount, collect, cap);

  select_kernel<<<dim3(1), dim3(BLOCK), 0, stream>>>(collect, ccount, cap, selB, selR, newbuf);
}